// UniLSTM_45071386804607
// MI455X (gfx1250) — compile-verified
//
#include <hip/hip_runtime.h>
#include <math.h>

#define T_ 2048
#define B_ 64
#define E_ 256
#define H_ 256
#define G_ 1024   // 4*H

typedef __attribute__((ext_vector_type(2))) float v2f;
typedef __attribute__((ext_vector_type(8))) float v8f;

__device__ __forceinline__ float sigm(float x) {
    return 1.0f / (1.0f + __expf(-x));
}

__device__ __forceinline__ float fast_tanh(float x) {
#if __has_builtin(__builtin_amdgcn_tanhf)
    return __builtin_amdgcn_tanhf(x);     // CDNA5 v_tanh_f32 (TRANS op)
#else
    float t = __expf(2.0f * x);
    return (t - 1.0f) / (t + 1.0f);
#endif
}

// ---------------------------------------------------------------------------
// init: zero the per-timestep barrier counters (runs every launch -> graph-safe)
// ---------------------------------------------------------------------------
__global__ void lstm_init_counters(unsigned int* __restrict__ cnt) {
    int i = blockIdx.x * blockDim.x + threadIdx.x;
    if (i < T_) cnt[i] = 0u;
}

// ---------------------------------------------------------------------------
// Phase 1: wi_all[t,b,g] = x[t,b,:] @ W_ih[:,g] + bias[g]
// M = T*B = 131072, N = 1024, K = 256.  fp32 WMMA 16x16x4.
// Workgroup (256 thr / 8 waves) computes 64 rows x 128 cols.
// ---------------------------------------------------------------------------
__global__ __launch_bounds__(256) void lstm_wi_gemm(
    const float* __restrict__ x, const float* __restrict__ w_ih,
    const float* __restrict__ bias_f, const float* __restrict__ bias_iog,
    float* __restrict__ wi)
{
    __shared__ float Wl[64 * 136];   // K-chunk 64 x 128 cols, pad 136 (bank-clean)
    __shared__ float Al[64 * 68];    // 64 rows x K-chunk 64, pad 68 (bank-clean)

    const int tid  = threadIdx.x;
    const int lane = tid & 31;
    const int wave = tid >> 5;
    const int ln   = lane & 15;
    const int hi   = lane >> 4;

    const int m0 = blockIdx.x * 64;     // 2048 M-blocks
    const int cb = blockIdx.y * 128;    // 8 N-blocks

    // accumulator init = bias[col] (C layout: fixed column per lane)
    const int col = cb + wave * 16 + ln;
    const float bv = (col < H_) ? bias_f[col] : bias_iog[col - H_];
    v8f acc[4];
    #pragma unroll
    for (int mt = 0; mt < 4; ++mt)
        #pragma unroll
        for (int r = 0; r < 8; ++r) acc[mt][r] = bv;

    for (int kc = 0; kc < 4; ++kc) {
        const int k0 = kc * 64;
        // stage W_ih chunk: 64 K-rows x 128 cols
        #pragma unroll
        for (int it = 0; it < 8; ++it) {
            int idx = tid + it * 256;          // 2048 float4
            int row = idx >> 5;                // 32 float4 per row
            int c4  = idx & 31;
            const float4 v = *(const float4*)&w_ih[(size_t)(k0 + row) * G_ + cb + c4 * 4];
            float* d = &Wl[row * 136 + c4 * 4];
            d[0] = v.x; d[1] = v.y; d[2] = v.z; d[3] = v.w;
        }
        // stage x chunk: 64 rows x 64 K
        #pragma unroll
        for (int it = 0; it < 4; ++it) {
            int idx = tid + it * 256;          // 1024 float4
            int row = idx >> 4;                // 16 float4 per row
            int c4  = idx & 15;
            const float4 v = *(const float4*)&x[(size_t)(m0 + row) * E_ + k0 + c4 * 4];
            float* d = &Al[row * 68 + c4 * 4];
            d[0] = v.x; d[1] = v.y; d[2] = v.z; d[3] = v.w;
        }
        __syncthreads();

        #pragma unroll 4
        for (int ks = 0; ks < 16; ++ks) {
            const int k = ks * 4 + 2 * hi;     // A/B lane K-pair per WMMA layout
            v2f bf;
            bf.x = Wl[k * 136 + wave * 16 + ln];
            bf.y = Wl[(k + 1) * 136 + wave * 16 + ln];
            #pragma unroll
            for (int mt = 0; mt < 4; ++mt) {
                const float* ap = &Al[(mt * 16 + ln) * 68 + k];
                v2f af; af.x = ap[0]; af.y = ap[1];
                acc[mt] = __builtin_amdgcn_wmma_f32_16x16x4_f32(
                    false, af, false, bf, (short)0, acc[mt], false, false);
            }
        }
        __syncthreads();
    }

    // write out (C layout: VGPR r -> row r + 8*hi, col = ln)
    #pragma unroll
    for (int mt = 0; mt < 4; ++mt) {
        #pragma unroll
        for (int r = 0; r < 8; ++r) {
            int row = m0 + mt * 16 + r + 8 * hi;
            wi[(size_t)row * G_ + col] = acc[mt][r];
        }
    }
}

// ---------------------------------------------------------------------------
// Phase 2: persistent sequential scan. 16 WGs x 512 thr (16 waves).
// WG g owns hidden units [16g, 16g+16) -> gate columns {j, 256+j, 512+j, 768+j}.
// W_hh slice + full h + cell slice resident in LDS (~158 KB).
// wi[t]/mask loads are issued BEFORE the GEMM so HBM latency hides behind WMMAs.
// Inter-WG sync: per-step release/acquire counter; h double-buffered in global.
// ---------------------------------------------------------------------------
__global__ __launch_bounds__(512) void lstm_scan(
    const float* __restrict__ mask, const float* __restrict__ w_hh,
    const float* __restrict__ wi, float* __restrict__ hglob,
    unsigned int* __restrict__ cnt, float* __restrict__ out)
{
    __shared__ float Whh[256 * 72];  // [k][cc], cc = gate*16 + jj, pad 72
    __shared__ float Hl[64 * 260];   // full h, [b][j], pad 260
    __shared__ float Pre[64 * 68];   // pre-activations, [b][cc], pad 68
    __shared__ float Cl[64 * 16];    // cell-state slice [b][jj]

    const int tid  = threadIdx.x;
    const int lane = tid & 31;
    const int wave = tid >> 5;
    const int ln   = lane & 15;
    const int hi   = lane >> 4;
    const int wg   = blockIdx.x;     // 0..15
    const int j0   = wg * 16;
    const int mt   = wave >> 2;      // M-subtile (batch rows)
    const int gi   = wave & 3;       // gate index

    // this thread's two gate-update elements (fixed for all t)
    const int b0  = tid >> 4,         jj0 = tid & 15,         j_0 = j0 + jj0;
    const int b1  = (tid + 512) >> 4, jj1 = (tid + 512) & 15, j_1 = j0 + jj1;

    // stage W_hh slice (once, reused for all 2048 steps)
    for (int idx = tid; idx < 256 * 64; idx += 512) {
        int k = idx >> 6, cc = idx & 63;
        int g = (cc >> 4) * H_ + j0 + (cc & 15);
        Whh[k * 72 + cc] = w_hh[(size_t)k * G_ + g];
    }
    for (int idx = tid; idx < 64 * 260; idx += 512) Hl[idx] = 0.0f;
    for (int idx = tid; idx < 64 * 16;  idx += 512) Cl[idx] = 0.0f;
    __syncthreads();

    for (int t = 0; t < T_; ++t) {
        // --- prefetch this step's wi / mask (consumed after the GEMM) ---
        const float* wit = &wi[(size_t)t * (B_ * G_)];
        float wf0 = wit[b0 * G_ +         j_0];
        float wi0 = wit[b0 * G_ +   H_ +  j_0];
        float wo0 = wit[b0 * G_ + 2*H_ +  j_0];
        float wg0 = wit[b0 * G_ + 3*H_ +  j_0];
        float wf1 = wit[b1 * G_ +         j_1];
        float wi1 = wit[b1 * G_ +   H_ +  j_1];
        float wo1 = wit[b1 * G_ + 2*H_ +  j_1];
        float wg1 = wit[b1 * G_ + 3*H_ +  j_1];
        float m0v = mask[b0 * T_ + t];
        float m1v = mask[b1 * T_ + t];

        // --- GEMM: pre = h @ W_hh[:, owned cols], K = 256 ---
        v8f a0 = {}, a1 = {}, a2 = {}, a3 = {};   // 4-way accumulator pipelining
        #pragma unroll 2
        for (int ks = 0; ks < 64; ks += 4) {
            #pragma unroll
            for (int u = 0; u < 4; ++u) {
                const int k = (ks + u) * 4 + 2 * hi;
                v2f bf; bf.x = Whh[k * 72 + gi * 16 + ln];
                        bf.y = Whh[(k + 1) * 72 + gi * 16 + ln];
                const float* ap = &Hl[(mt * 16 + ln) * 260 + k];
                v2f af; af.x = ap[0]; af.y = ap[1];
                v8f& a = (u == 0) ? a0 : (u == 1) ? a1 : (u == 2) ? a2 : a3;
                a = __builtin_amdgcn_wmma_f32_16x16x4_f32(
                    false, af, false, bf, (short)0, a, false, false);
            }
        }
        v8f acc = (a0 + a1) + (a2 + a3);
        #pragma unroll
        for (int r = 0; r < 8; ++r)
            Pre[(mt * 16 + r + 8 * hi) * 68 + gi * 16 + ln] = acc[r];
        __syncthreads();

        // --- gates / state update: 2 elements per thread ---
        {
            float pf = Pre[b0 * 68 +  0 + jj0] + wf0;
            float pi = Pre[b0 * 68 + 16 + jj0] + wi0;
            float po = Pre[b0 * 68 + 32 + jj0] + wo0;
            float pg = Pre[b0 * 68 + 48 + jj0] + wg0;
            float c_old = Cl[b0 * 16 + jj0];
            float h_old = Hl[b0 * 260 + j_0];
            float c1 = sigm(pf) * c_old + sigm(pi) * fast_tanh(pg);
            c1 = c1 * m0v + c_old * (1.0f - m0v);
            float h1 = sigm(po) * fast_tanh(c1);
            h1 = h1 * m0v + h_old * (1.0f - m0v);
            Cl[b0 * 16 + jj0] = c1;
            out[(size_t)t * (B_ * H_) + b0 * H_ + j_0] = h1 * m0v;
            hglob[(t & 1) * (B_ * H_) + b0 * H_ + j_0] = h1;
            if (t == T_ - 1) {
                out[(size_t)T_ * B_ * H_ +           b0 * H_ + j_0] = h1;
                out[(size_t)T_ * B_ * H_ + B_ * H_ + b0 * H_ + j_0] = c1;
            }
        }
        {
            float pf = Pre[b1 * 68 +  0 + jj1] + wf1;
            float pi = Pre[b1 * 68 + 16 + jj1] + wi1;
            float po = Pre[b1 * 68 + 32 + jj1] + wo1;
            float pg = Pre[b1 * 68 + 48 + jj1] + wg1;
            float c_old = Cl[b1 * 16 + jj1];
            float h_old = Hl[b1 * 260 + j_1];
            float c1 = sigm(pf) * c_old + sigm(pi) * fast_tanh(pg);
            c1 = c1 * m1v + c_old * (1.0f - m1v);
            float h1 = sigm(po) * fast_tanh(c1);
            h1 = h1 * m1v + h_old * (1.0f - m1v);
            Cl[b1 * 16 + jj1] = c1;
            out[(size_t)t * (B_ * H_) + b1 * H_ + j_1] = h1 * m1v;
            hglob[(t & 1) * (B_ * H_) + b1 * H_ + j_1] = h1;
            if (t == T_ - 1) {
                out[(size_t)T_ * B_ * H_ +           b1 * H_ + j_1] = h1;
                out[(size_t)T_ * B_ * H_ + B_ * H_ + b1 * H_ + j_1] = c1;
            }
        }
        __threadfence();
        __syncthreads();

        // --- cross-WG barrier for step t ---
        if (tid == 0) {
            __hip_atomic_fetch_add(&cnt[t], 1u, __ATOMIC_ACQ_REL,
                                   __HIP_MEMORY_SCOPE_AGENT);
            while (__hip_atomic_load(&cnt[t], __ATOMIC_ACQUIRE,
                                     __HIP_MEMORY_SCOPE_AGENT) < 16u) {
                __builtin_amdgcn_s_sleep(1);
            }
            __threadfence();
        }
        __syncthreads();

        // --- reload full h_t into LDS ---
        const float4* hg4 = (const float4*)&hglob[(t & 1) * (B_ * H_)];
        for (int idx = tid; idx < 64 * 64; idx += 512) {
            int row = idx >> 6, c4 = idx & 63;
            float4 v = hg4[row * 64 + c4];
            float* d = &Hl[row * 260 + c4 * 4];
            d[0] = v.x; d[1] = v.y; d[2] = v.z; d[3] = v.w;
        }
        __syncthreads();
    }
}

// ---------------------------------------------------------------------------
extern "C" void kernel_launch(void* const* d_in, const int* in_sizes, int n_in,
                              void* d_out, int out_size, void* d_ws, size_t ws_size,
                              hipStream_t stream) {
    const float* x        = (const float*)d_in[0];
    const float* mask     = (const float*)d_in[1];
    const float* w_ih     = (const float*)d_in[2];
    const float* w_hh     = (const float*)d_in[3];
    const float* bias_f   = (const float*)d_in[4];
    const float* bias_iog = (const float*)d_in[5];
    float* out = (float*)d_out;

    char* ws = (char*)d_ws;
    unsigned int* cnt = (unsigned int*)(ws);                 // 2048 u32
    float* hglob      = (float*)(ws + 8192);                 // 2 x 64x256 f32
    float* wi         = (float*)(ws + 262144);               // T*B*G f32 (512 MB)

    lstm_init_counters<<<dim3((T_ + 255) / 256), dim3(256), 0, stream>>>(cnt);
    lstm_wi_gemm<<<dim3(2048, 8), dim3(256), 0, stream>>>(x, w_ih, bias_f, bias_iog, wi);
    lstm_scan<<<dim3(16), dim3(512), 0, stream>>>(mask, w_hh, wi, hglob, cnt, out);
}